// LightGCN_12627203851097
// MI455X (gfx1250) — compile-verified
//
#include <hip/hip_runtime.h>
#include <math.h>

#define NUSR 100000
#define NITM 100000
#define NAUT 10000
#define NG0  (NUSR + NITM)   // 200000
#define NG1  (NUSR + NAUT)   // 110000
#define DIM  64

typedef __attribute__((ext_vector_type(2))) float v2f;
typedef __attribute__((ext_vector_type(8))) float v8f;

// ---------------------------------------------------------------- zero fill
__global__ void fill0_kernel(float4* __restrict__ p, long n4) {
    long i  = (long)blockIdx.x * blockDim.x + threadIdx.x;
    long st = (long)gridDim.x * blockDim.x;
    float4 z{0.f, 0.f, 0.f, 0.f};
    for (; i < n4; i += st) p[i] = z;
}

// ------------------------------------------- init: cur = acc = [embA; embB]
// acc is split across two non-contiguous d_out slots (accA rows [0,na), accB rest)
__global__ void init_branch_kernel(const float4* __restrict__ ea, long na4,
                                   const float4* __restrict__ eb, long nb4,
                                   float4* __restrict__ cur,
                                   float4* __restrict__ accA,
                                   float4* __restrict__ accB) {
    long i  = (long)blockIdx.x * blockDim.x + threadIdx.x;
    long st = (long)gridDim.x * blockDim.x;
    long n  = na4 + nb4;
    for (; i < n; i += st) {
        float4 v;
        if (i < na4) { v = ea[i]; accA[i] = v; }
        else         { v = eb[i - na4]; accB[i - na4] = v; }
        cur[i] = v;
    }
}

// ---------------------------------------------------- acc(split) += next
__global__ void acc_add_split_kernel(const float4* __restrict__ nxt, long na4, long ntot4,
                                     float4* __restrict__ accA, float4* __restrict__ accB) {
    long i  = (long)blockIdx.x * blockDim.x + threadIdx.x;
    long st = (long)gridDim.x * blockDim.x;
    for (; i < ntot4; i += st) {
        float4 v = nxt[i];
        float4* d = (i < na4) ? (accA + i) : (accB + (i - na4));
        float4 o = *d;
        o.x += v.x; o.y += v.y; o.z += v.z; o.w += v.w;
        *d = o;
    }
}

// ---------------------------------------------------------- p *= s (in place)
__global__ void scale_kernel(float4* __restrict__ p, long n4, float s) {
    long i  = (long)blockIdx.x * blockDim.x + threadIdx.x;
    long st = (long)gridDim.x * blockDim.x;
    for (; i < n4; i += st) {
        float4 v = p[i];
        v.x *= s; v.y *= s; v.z *= s; v.w *= s;
        p[i] = v;
    }
}

// --------------------------------------------------------------------- SpMM
// y[row] += val * x[col]; one wave per edge iteration: 32 lanes load float2
// (256B coalesced row gather), per-lane global_atomic_add_f32 scatter.
// Edge metadata loaded coalesced (32 edges/lane-vector) with NON-TEMPORAL
// hints: the read-once row/col/val stream (77MB per g0 sweep) must not evict
// the feature/accumulator arrays (~51MB + ~51MB), which fit in the 192MB L2
// and serve the random gathers/atomics at L2 bandwidth. Broadcast via shfl;
// global_prefetch_b8 issued 8 edges ahead hides random-gather latency.
__global__ void spmm_kernel(const int* __restrict__ erow, const int* __restrict__ ecol,
                            const float* __restrict__ eval, int nE,
                            const float* __restrict__ x, float* __restrict__ y) {
    const int lane    = threadIdx.x & 31;
    const int wave    = blockIdx.x * (blockDim.x >> 5) + (threadIdx.x >> 5);
    const int nWaves  = gridDim.x * (blockDim.x >> 5);
    for (int base = wave * 32; base < nE; base += nWaves * 32) {
        int e = base + lane;
        int er = 0, ec = 0; float ev = 0.f;
        if (e < nE) {
            er = __builtin_nontemporal_load(erow + e);   // TH=NT: streaming, read-once
            ec = __builtin_nontemporal_load(ecol + e);
            ev = __builtin_nontemporal_load(eval + e);
        }
        int cnt = nE - base; if (cnt > 32) cnt = 32;
        for (int j = 0; j < cnt; ++j) {
            if (j + 8 < cnt) {   // prefetch row 8 edges ahead (global_prefetch_b8)
                int cp = __shfl(ec, j + 8, 32);
                __builtin_prefetch(x + (size_t)cp * DIM + (lane << 1), 0, 0);
            }
            int   rj = __shfl(er, j, 32);
            int   cj = __shfl(ec, j, 32);
            float vj = __shfl(ev, j, 32);
            const float2 xv = *(const float2*)(x + (size_t)cj * DIM + (lane << 1));
            float* dst = y + (size_t)rj * DIM + (lane << 1);
            atomicAdd(dst + 0, xv.x * vj);
            atomicAdd(dst + 1, xv.y * vj);
        }
    }
}

// ----------------------------------------------------- WMMA row-normalize
// Per 16-row tile: D = (X .* X) * ones  via chained V_WMMA_F32_16X16X4_F32.
// Every column of D broadcasts the row norm^2. Then rescale rows.
// mix==0: out = x / max(||x||,eps)
// mix==1: out = 0.5 * x/max(||x||,eps) + 0.5 * out   (in-place blend)
__global__ void normalize16_kernel(const float* __restrict__ x, float* __restrict__ out,
                                   int ngroups, int mix) {
    __shared__ float shn[8 * 16];               // 8 waves * 16 row-norms
    const int lane  = threadIdx.x & 31;
    const int wid   = threadIdx.x >> 5;
    const int half  = lane >> 4;                // 0: K={0,1}, 1: K={2,3}
    const int m     = lane & 15;                // A-matrix row
    const int wstep = gridDim.x * (blockDim.x >> 5);
    for (int g = blockIdx.x * (blockDim.x >> 5) + wid; g < ngroups; g += wstep) {
        const float* xg = x + (size_t)g * 16 * DIM;
        v8f c = {};
        v2f b; b.x = 1.0f; b.y = 1.0f;          // ones matrix (layout-invariant)
#pragma unroll
        for (int kk = 0; kk < 16; ++kk) {       // K = 64 in chunks of 4
            const float2 xv = *(const float2*)(xg + (size_t)m * DIM + (kk << 2) + (half << 1));
            v2f a; a.x = xv.x * xv.x; a.y = xv.y * xv.y;
            c = __builtin_amdgcn_wmma_f32_16x16x4_f32(false, a, false, b,
                                                      (short)0, c, false, false);
        }
        // lanes 0 / 16 hold norms of rows 0..7 / 8..15 (any column of D)
        if (m == 0) {
#pragma unroll
            for (int v = 0; v < 8; ++v) shn[wid * 16 + half * 8 + v] = c[v];
        }
        // same-wave DS ops are in-order: RAW through LDS is safe without barrier
        float n2 = shn[wid * 16 + m];
        float s  = 1.0f / fmaxf(sqrtf(n2), 1e-12f);
        const float4* src = (const float4*)(xg + (size_t)m * DIM + half * 32);
        float4*       dst = (float4*)(out + ((size_t)g * 16 + m) * DIM + half * 32);
        if (mix) {
#pragma unroll
            for (int q = 0; q < 8; ++q) {
                float4 t = src[q]; float4 o = dst[q];
                o.x = 0.5f * (s * t.x) + 0.5f * o.x;
                o.y = 0.5f * (s * t.y) + 0.5f * o.y;
                o.z = 0.5f * (s * t.z) + 0.5f * o.z;
                o.w = 0.5f * (s * t.w) + 0.5f * o.w;
                dst[q] = o;
            }
        } else {
#pragma unroll
            for (int q = 0; q < 8; ++q) {
                float4 t = src[q];
                t.x *= s; t.y *= s; t.z *= s; t.w *= s;
                dst[q] = t;
            }
        }
    }
}

// ===========================================================================
extern "C" void kernel_launch(void* const* d_in, const int* in_sizes, int n_in,
                              void* d_out, int out_size, void* d_ws, size_t ws_size,
                              hipStream_t stream) {
    const float* emb_user   = (const float*)d_in[0];
    const float* emb_item   = (const float*)d_in[1];
    const float* emb_author = (const float*)d_in[2];
    const int*   g0r = (const int*)d_in[3];  const int* g0c = (const int*)d_in[4];
    const float* g0v = (const float*)d_in[5];
    const int*   g1r = (const int*)d_in[6];  const int* g1c = (const int*)d_in[7];
    const float* g1v = (const float*)d_in[8];
    const int*   g3r = (const int*)d_in[9];  const int* g3c = (const int*)d_in[10];
    const float* g3v = (const float*)d_in[11];
    const int*   g4r = (const int*)d_in[12]; const int* g4c = (const int*)d_in[13];
    const float* g4v = (const float*)d_in[14];
    const int*   g5r = (const int*)d_in[15]; const int* g5c = (const int*)d_in[16];
    const float* g5v = (const float*)d_in[17];
    const int E0 = in_sizes[3], E1 = in_sizes[6], E3 = in_sizes[9],
              E4 = in_sizes[12], E5 = in_sizes[15];

    float* out      = (float*)d_out;
    float* out_au   = out;                                        // atom_users
    float* out_nau  = out + (size_t)NUSR * DIM;                   // non_atom_users
    float* out_ai   = out + (size_t)2 * NUSR * DIM;               // atom_items
    float* out_nai  = out + ((size_t)2 * NUSR + NITM) * DIM;      // non_atom_items
    float* out_aa   = out + ((size_t)2 * NUSR + 2 * NITM) * DIM;  // atom_authors
    float* out_naa  = out + ((size_t)2 * NUSR + 2 * NITM + NAUT) * DIM; // non_atom_authors

    float* b1 = (float*)d_ws;                 // N0*64 floats
    float* b2 = b1 + (size_t)NG0 * DIM;       // N0*64 floats

    const int TB = 256;
    auto nblk = [](long n4) { long b = (n4 + 255) / 256; return (int)(b > 4096 ? 4096 : b); };
#define ZERO(ptr, nfl) fill0_kernel<<<nblk((nfl) / 4), TB, 0, stream>>>((float4*)(ptr), (long)(nfl) / 4)
#define SPMM(r, c, v, ne, xx, yy) spmm_kernel<<<2048, TB, 0, stream>>>((r), (c), (v), (ne), (xx), (yy))

    // ======================= atom branch: user-item graph (g0, N0 nodes) ===
    init_branch_kernel<<<nblk((long)NG0 * 16), TB, 0, stream>>>(
        (const float4*)emb_user, (long)NUSR * 16,
        (const float4*)emb_item, (long)NITM * 16,
        (float4*)b1, (float4*)out_au, (float4*)out_ai);
    for (int L = 0; L < 3; ++L) {
        float* cur = (L & 1) ? b2 : b1;
        float* nxt = (L & 1) ? b1 : b2;
        ZERO(nxt, (long)NG0 * DIM);
        SPMM(g0r, g0c, g0v, E0, cur, nxt);
        acc_add_split_kernel<<<nblk((long)NG0 * 16), TB, 0, stream>>>(
            (const float4*)nxt, (long)NUSR * 16, (long)NG0 * 16,
            (float4*)out_au, (float4*)out_ai);
    }
    scale_kernel<<<nblk((long)NUSR * 16), TB, 0, stream>>>((float4*)out_au, (long)NUSR * 16, 0.25f);
    scale_kernel<<<nblk((long)NITM * 16), TB, 0, stream>>>((float4*)out_ai, (long)NITM * 16, 0.25f);

    // atom_authors = normalize(g3 @ items0)         (t3 in b1, free now)
    ZERO(b1, (long)NAUT * DIM);
    SPMM(g3r, g3c, g3v, E3, out_ai, b1);
    normalize16_kernel<<<(NAUT / 16 + 7) / 8, TB, 0, stream>>>(b1, out_aa, NAUT / 16, 0);

    // atom_items = 0.5*normalize(g5 @ items0) + 0.5*items0   (t5 in b2)
    ZERO(b2, (long)NITM * DIM);
    SPMM(g5r, g5c, g5v, E5, out_ai, b2);
    normalize16_kernel<<<512, TB, 0, stream>>>(b2, out_ai, NITM / 16, 1);

    // ==================== non-atom branch: user-author graph (g1, N1) ======
    init_branch_kernel<<<nblk((long)NG1 * 16), TB, 0, stream>>>(
        (const float4*)emb_user,   (long)NUSR * 16,
        (const float4*)emb_author, (long)NAUT * 16,
        (float4*)b1, (float4*)out_nau, (float4*)out_naa);
    for (int L = 0; L < 3; ++L) {
        float* cur = (L & 1) ? b2 : b1;
        float* nxt = (L & 1) ? b1 : b2;
        ZERO(nxt, (long)NG1 * DIM);
        SPMM(g1r, g1c, g1v, E1, cur, nxt);
        acc_add_split_kernel<<<nblk((long)NG1 * 16), TB, 0, stream>>>(
            (const float4*)nxt, (long)NUSR * 16, (long)NG1 * 16,
            (float4*)out_nau, (float4*)out_naa);
    }
    scale_kernel<<<nblk((long)NUSR * 16), TB, 0, stream>>>((float4*)out_nau, (long)NUSR * 16, 0.25f);
    scale_kernel<<<nblk((long)NAUT * 16), TB, 0, stream>>>((float4*)out_naa, (long)NAUT * 16, 0.25f);

    // non_atom_items = normalize(g4 @ non_atom_authors)   (t4 in b2)
    ZERO(b2, (long)NITM * DIM);
    SPMM(g4r, g4c, g4v, E4, out_naa, b2);
    normalize16_kernel<<<512, TB, 0, stream>>>(b2, out_nai, NITM / 16, 0);

#undef ZERO
#undef SPMM
    (void)n_in; (void)out_size; (void)ws_size; (void)E1;
}